// ModelNew_17514876633415
// MI455X (gfx1250) — compile-verified
//
#include <hip/hip_runtime.h>
#include <hip/hip_bf16.h>

// GRU dims (fixed by the reference)
#define TT_ 512
#define BB_ 64
#define II_ 512
#define HH_ 512
#define LL_ 3

typedef __attribute__((ext_vector_type(2))) float v2f;
typedef __attribute__((ext_vector_type(8))) float v8f;

__device__ __forceinline__ float sigmoid_f(float x) {
    return 1.0f / (1.0f + __expf(-x));
}
__device__ __forceinline__ float tanh_f(float x) {
    float e = __expf(-2.0f * x);
    return (1.0f - e) / (1.0f + e);
}

__device__ __forceinline__ v8f wmma_f32(v2f a, v2f b, v8f c) {
    return __builtin_amdgcn_wmma_f32_16x16x4_f32(
        false, a, false, b, (short)0, c, false, false);
}

// ---------------------------------------------------------------------------
// Kernel 1: gih[d, t*B+b, g] = sum_k x[t,b,k] * w_ih[g,k] + b_ih[g]
// Register-blocked: each wave owns a 32x64 output tile (2 M-frags x 4 N-frags,
// 8 fp32 WMMA accumulators) so 6 fragment loads feed 8 WMMAs per K-step.
// M = T*B = 32768 rows, N = 3H = 1536, K = in_sz (512 or 1024).
// grid.x = (M/32 * N/64) / 8 waves-per-block = 3072, grid.y = direction.
// ---------------------------------------------------------------------------
__global__ void __launch_bounds__(256)
gih_gemm_kernel(const float* __restrict__ x,
                const float* __restrict__ wih_f, const float* __restrict__ bih_f,
                const float* __restrict__ wih_b, const float* __restrict__ bih_b,
                float* __restrict__ gih, int K)
{
    const int dir  = blockIdx.y;
    const int wave = threadIdx.x >> 5;
    const int tid  = blockIdx.x * 8 + wave;           // 0 .. 1024*24-1
    const int lane = threadIdx.x & 31;
    const int half = lane >> 4;                        // 0 or 1
    const int lm   = lane & 15;                        // 0..15

    const int NT = (3 * HH_) / 64;                     // 24 N strips
    const int m0 = (tid / NT) * 32;
    const int g0 = (tid % NT) * 64;

    const float* __restrict__ w    = dir ? wih_b : wih_f;
    const float* __restrict__ bias = dir ? bih_b : bih_f;

    // A fragments: rows m0+lm and m0+16+lm of x; K-contiguous pairs.
    const float* ap0 = x + (size_t)(m0 + lm) * K + 2 * half;
    const float* ap1 = x + (size_t)(m0 + 16 + lm) * K + 2 * half;
    // B fragments: rows g0+16*nn+lm of w_ih (output uses W^T).
    const float* bp0 = w + (size_t)(g0 + lm) * K + 2 * half;
    const float* bp1 = w + (size_t)(g0 + 16 + lm) * K + 2 * half;
    const float* bp2 = w + (size_t)(g0 + 32 + lm) * K + 2 * half;
    const float* bp3 = w + (size_t)(g0 + 48 + lm) * K + 2 * half;

    v8f acc[2][4];
#pragma unroll
    for (int mm = 0; mm < 2; ++mm)
#pragma unroll
        for (int nn = 0; nn < 4; ++nn)
            acc[mm][nn] = (v8f){};

    for (int k0 = 0; k0 < K; k0 += 4) {
        v2f a0 = *(const v2f*)(ap0 + k0);
        v2f a1 = *(const v2f*)(ap1 + k0);
        v2f b0 = *(const v2f*)(bp0 + k0);
        v2f b1 = *(const v2f*)(bp1 + k0);
        v2f b2 = *(const v2f*)(bp2 + k0);
        v2f b3 = *(const v2f*)(bp3 + k0);
        acc[0][0] = wmma_f32(a0, b0, acc[0][0]);
        acc[0][1] = wmma_f32(a0, b1, acc[0][1]);
        acc[0][2] = wmma_f32(a0, b2, acc[0][2]);
        acc[0][3] = wmma_f32(a0, b3, acc[0][3]);
        acc[1][0] = wmma_f32(a1, b0, acc[1][0]);
        acc[1][1] = wmma_f32(a1, b1, acc[1][1]);
        acc[1][2] = wmma_f32(a1, b2, acc[1][2]);
        acc[1][3] = wmma_f32(a1, b3, acc[1][3]);
    }

    float* __restrict__ out = gih + (size_t)dir * TT_ * BB_ * 3 * HH_;
#pragma unroll
    for (int nn = 0; nn < 4; ++nn) {
        const float bv = bias[g0 + 16 * nn + lm];
#pragma unroll
        for (int mm = 0; mm < 2; ++mm) {
#pragma unroll
            for (int v = 0; v < 8; ++v) {
                int m = m0 + 16 * mm + v + 8 * half;   // global row
                out[(size_t)m * (3 * HH_) + g0 + 16 * nn + lm] = acc[mm][nn][v] + bv;
            }
        }
    }
}

// ---------------------------------------------------------------------------
// Kernel 2: copy initial hidden state (both directions of one layer) into the
// ping-pong buffer slot 0.  h0slice points at h0[2*layer], shape (2, B, H).
// ---------------------------------------------------------------------------
__global__ void __launch_bounds__(256)
init_h_kernel(const float* __restrict__ h0slice, float* __restrict__ hbuf)
{
    int idx = blockIdx.x * blockDim.x + threadIdx.x;   // 0 .. 2*B*H-1
    int dir = idx / (BB_ * HH_);
    int i   = idx % (BB_ * HH_);
    hbuf[(size_t)dir * 2 * BB_ * HH_ + i] = h0slice[idx];
}

// ---------------------------------------------------------------------------
// Kernel 3: one GRU time step, both directions.
// Each wave owns one (16 batch x 16 hidden) tile and computes r/z/n gate
// tiles with shared A (h) fragments, then the fused gate math.
// 2 dirs * 4 btiles * 32 gtiles = 256 waves = 32 blocks of 256 threads.
// ---------------------------------------------------------------------------
__global__ void __launch_bounds__(256)
gru_step_kernel(const float* __restrict__ whh_f, const float* __restrict__ bhh_f,
                const float* __restrict__ whh_b, const float* __restrict__ bhh_b,
                const float* __restrict__ gih,
                float* __restrict__ hbuf,
                float* __restrict__ out,           // (T, B, 2H), dir selects half
                int t)
{
    const int wid  = blockIdx.x * 8 + (threadIdx.x >> 5);  // 0..255
    const int dir  = wid >> 7;
    const int rem  = wid & 127;
    const int btile = rem >> 5;                            // 0..3
    const int gtile = rem & 31;                            // 0..31
    const int lane = threadIdx.x & 31;
    const int half = lane >> 4;
    const int lm   = lane & 15;

    const int b0 = btile * 16;
    const int j0 = gtile * 16;
    const int tt = dir ? (TT_ - 1 - t) : t;

    const float* __restrict__ whh = dir ? whh_b : whh_f;
    const float* __restrict__ bhh = dir ? bhh_b : bhh_f;

    float* hbase = hbuf + (size_t)dir * 2 * BB_ * HH_;
    const float* __restrict__ h_in  = hbase + (size_t)(t & 1) * BB_ * HH_;
    float* __restrict__       h_out = hbase + (size_t)((t + 1) & 1) * BB_ * HH_;

    // ghh[b, g] = sum_k h[b,k] * whh[g,k]   (g in {j, H+j, 2H+j})
    const float* aptr = h_in + (size_t)(b0 + lm) * HH_ + 2 * half;
    const float* brp  = whh + (size_t)(j0 + lm) * HH_ + 2 * half;
    const float* bzp  = whh + (size_t)(HH_ + j0 + lm) * HH_ + 2 * half;
    const float* bnp  = whh + (size_t)(2 * HH_ + j0 + lm) * HH_ + 2 * half;

    v8f acc_r = {}, acc_z = {}, acc_n = {};
    for (int k0 = 0; k0 < HH_; k0 += 4) {
        v2f a  = *(const v2f*)(aptr + k0);
        v2f br = *(const v2f*)(brp + k0);
        v2f bz = *(const v2f*)(bzp + k0);
        v2f bn = *(const v2f*)(bnp + k0);
        acc_r = wmma_f32(a, br, acc_r);
        acc_z = wmma_f32(a, bz, acc_z);
        acc_n = wmma_f32(a, bn, acc_n);
    }

    const float* __restrict__ gbase =
        gih + (size_t)dir * TT_ * BB_ * 3 * HH_ + (size_t)tt * BB_ * 3 * HH_;

#pragma unroll
    for (int v = 0; v < 8; ++v) {
        const int m  = v + 8 * half;
        const int bb = b0 + m;
        const int j  = j0 + lm;
        const float* grow = gbase + (size_t)bb * (3 * HH_);

        float r  = sigmoid_f(grow[j]            + acc_r[v] + bhh[j]);
        float z  = sigmoid_f(grow[HH_ + j]      + acc_z[v] + bhh[HH_ + j]);
        float nn = tanh_f   (grow[2 * HH_ + j]  + r * (acc_n[v] + bhh[2 * HH_ + j]));
        float hp = h_in[(size_t)bb * HH_ + j];
        float hv = (1.0f - z) * nn + z * hp;

        h_out[(size_t)bb * HH_ + j] = hv;
        out[((size_t)tt * BB_ + bb) * (2 * HH_) + (size_t)dir * HH_ + j] = hv;
    }
}

// ---------------------------------------------------------------------------
extern "C" void kernel_launch(void* const* d_in, const int* in_sizes, int n_in,
                              void* d_out, int out_size, void* d_ws, size_t ws_size,
                              hipStream_t stream) {
    const float* x  = (const float*)d_in[0];
    const float* h0 = (const float*)d_in[1];
    // weights: d_in[2 + layer*8 + i], i: 0=w_ih_f 1=w_hh_f 2=b_ih_f 3=b_hh_f
    //                                    4=w_ih_b 5=w_hh_b 6=b_ih_b 7=b_hh_b
    auto W = [&](int layer, int i) -> const float* {
        return (const float*)d_in[2 + layer * 8 + i];
    };

    float* ws = (float*)d_ws;
    const size_t gihSz = (size_t)2 * TT_ * BB_ * 3 * HH_;   // both directions
    const size_t bufSz = (size_t)TT_ * BB_ * 2 * HH_;       // layer output
    float* gih  = ws;
    float* bufA = gih + gihSz;
    float* bufB = bufA + bufSz;
    float* hbuf = bufB + bufSz;                             // 2 dirs * 2 pp * B*H

    const int MT = (TT_ * BB_) / 32;                        // 1024 M strips
    const int NT = (3 * HH_) / 64;                          // 24 N strips
    const int gihBlocks = (MT * NT) / 8;                    // 8 waves per block

    for (int layer = 0; layer < LL_; ++layer) {
        const float* inx = (layer == 0) ? x : ((layer == 1) ? bufA : bufB);
        const int K      = (layer == 0) ? II_ : 2 * HH_;
        float* outy      = (layer == 2) ? (float*)d_out
                                        : ((layer == 0) ? bufA : bufB);

        gih_gemm_kernel<<<dim3(gihBlocks, 2), 256, 0, stream>>>(
            inx, W(layer, 0), W(layer, 2), W(layer, 4), W(layer, 6), gih, K);

        init_h_kernel<<<(2 * BB_ * HH_) / 256, 256, 0, stream>>>(
            h0 + (size_t)2 * layer * BB_ * HH_, hbuf);

        for (int t = 0; t < TT_; ++t) {
            gru_step_kernel<<<32, 256, 0, stream>>>(
                W(layer, 1), W(layer, 3), W(layer, 5), W(layer, 7),
                gih, hbuf, outy, t);
        }
    }
}